// ArmaNet_bench_13271448944809
// MI455X (gfx1250) — compile-verified
//
#include <hip/hip_runtime.h>
#include <hip/hip_bf16.h>
#include <math.h>

// ---------------- constants (match reference) ----------------
#define KST 3        // num_stacks
#define HCH 16       // hidden channels
#define LIT 4        // ARMA iterations
#define BN_EPS 1e-5f

typedef float v2f __attribute__((ext_vector_type(2)));
typedef float v8f __attribute__((ext_vector_type(8)));

// Guaranteed-native non-returning fp32 scatter-add (ISA §15.18.3 GLOBAL_ATOMIC_ADD_F32,
// default TH -> non-returning). Avoids any CAS-loop expansion of atomicrmw fadd.
__device__ __forceinline__ void atomAddF(float* p, float v) {
    asm volatile("global_atomic_add_f32 %0, %1, off" :: "v"(p), "v"(v) : "memory");
}

// ---------------- generic zero ----------------
__global__ void zero_f32(float* __restrict__ p, int n) {
    int i = blockIdx.x * blockDim.x + threadIdx.x;
    int stride = gridDim.x * blockDim.x;
    for (; i < n; i += stride) p[i] = 0.0f;
}

// ---------------- gcn_norm ----------------
__global__ void deg_scatter(const int* __restrict__ ei, const float* __restrict__ ea,
                            float* __restrict__ deg, int E) {
    int e = blockIdx.x * blockDim.x + threadIdx.x;
    if (e >= E) return;
    atomAddF(&deg[ei[E + e]], ea[e]);   // segment_sum(edge_attr, col)
}

__global__ void compute_dinv(const float* __restrict__ deg, float* __restrict__ dinv, int N) {
    int i = blockIdx.x * blockDim.x + threadIdx.x;
    if (i >= N) return;
    float d = deg[i];
    dinv[i] = (d > 0.0f) ? rsqrtf(fmaxf(d, 1e-12f)) : 0.0f;
}

__global__ void compute_norm(const int* __restrict__ ei, const float* __restrict__ ea,
                             const float* __restrict__ dinv, float* __restrict__ norm, int E) {
    int e = blockIdx.x * blockDim.x + threadIdx.x;
    if (e >= E) return;
    norm[e] = dinv[ei[e]] * ea[e] * dinv[ei[E + e]];
}

// ---------------- ARMAConv 1 (Fin=1 -> H=16) ----------------
__global__ void conv1_init(const float* __restrict__ x, const float* __restrict__ w_init,
                           const float* __restrict__ w_root,
                           float* __restrict__ out, float* __restrict__ root, int N) {
    int idx = blockIdx.x * blockDim.x + threadIdx.x;     // over K*N
    if (idx >= KST * N) return;
    int k = idx / N, n = idx % N;
    float xv = x[n];
    size_t base = ((size_t)k * N + n) * HCH;
#pragma unroll
    for (int f = 0; f < HCH; ++f) {
        out[base + f]  = xv * w_init[k * HCH + f];
        root[base + f] = xv * w_root[k * HCH + f];
    }
}

// Fused: out = relu(acc + root + b1[k]) @ w1[k]  (4x chained V_WMMA_F32_16X16X4_F32),
// and acc is re-zeroed in the same pass for the next scatter.
// One wave per (stack, 16-node tile); wave-uniform exit keeps EXEC all-1s for WMMA.
__global__ __launch_bounds__(256) void arma1_fused(float* __restrict__ acc,
                                                   const float* __restrict__ root,
                                                   const float* __restrict__ b1,
                                                   const float* __restrict__ w1,
                                                   float* __restrict__ out,
                                                   int numTiles, int totalWaves) {
    int wave = blockIdx.x * (blockDim.x >> 5) + (threadIdx.x >> 5);
    if (wave >= totalWaves) return;
    int lane  = threadIdx.x & 31;
    int half  = lane >> 4;
    int col16 = lane & 15;
    int k    = wave / numTiles;
    int tile = wave % numTiles;

    const float* wk = w1 + k * (HCH * HCH);
    size_t tbase = ((size_t)k * numTiles + tile) * 16 * HCH;
    float* ap       = acc + tbase;     // [16 nodes][16 feats]
    const float* rp = root + tbase;
    float* op       = out + tbase;

    v8f c = {0.f, 0.f, 0.f, 0.f, 0.f, 0.f, 0.f, 0.f};
#pragma unroll
    for (int kk = 0; kk < 4; ++kk) {
        int fa = 4 * kk + 2 * half;                      // K-dim elements for this lane-half
        // A 16x4 fragment: lane m=col16 holds A[m][fa], A[m][fa+1] = relu(acc+root+b)
        float a0 = fmaxf(ap[col16 * HCH + fa]     + rp[col16 * HCH + fa]     + b1[k * HCH + fa],     0.f);
        float a1 = fmaxf(ap[col16 * HCH + fa + 1] + rp[col16 * HCH + fa + 1] + b1[k * HCH + fa + 1], 0.f);
        v2f a = {a0, a1};
        // B 4x16 fragment: lane n=col16 holds B[fa][n], B[fa+1][n] (B = w1[k])
        v2f b = {wk[fa * HCH + col16], wk[(fa + 1) * HCH + col16]};
        c = __builtin_amdgcn_wmma_f32_16x16x4_f32(false, a, false, b, (short)0, c, false, false);
    }
    // re-zero acc for the next scatter (lanes collectively cover the whole tile)
#pragma unroll
    for (int kk = 0; kk < 4; ++kk) {
        int fa = 4 * kk + 2 * half;
        ap[col16 * HCH + fa]     = 0.0f;
        ap[col16 * HCH + fa + 1] = 0.0f;
    }
    // D 16x16 f32: VGPR j, this lane -> row M = j + 8*half, col N = col16
#pragma unroll
    for (int j = 0; j < 8; ++j)
        op[(j + 8 * half) * HCH + col16] = c[j];
}

// message + scatter: acc[k, col[e], :] += out[k, row[e], :] * norm[e]
// thread per edge, loops over K stacks (indices/norm loaded once)
__global__ void scatter1(const float* __restrict__ out, const int* __restrict__ ei,
                         const float* __restrict__ norm, float* __restrict__ acc,
                         int E, int N) {
    int e = blockIdx.x * blockDim.x + threadIdx.x;
    if (e >= E) return;
    int r = ei[e], c = ei[E + e];
    float w = norm[e];
#pragma unroll
    for (int k = 0; k < KST; ++k) {
        const float4* src = (const float4*)(out + ((size_t)k * N + r) * HCH);
        float* dst = acc + ((size_t)k * N + c) * HCH;
#pragma unroll
        for (int q = 0; q < 4; ++q) {
            float4 v = src[q];
            atomAddF(dst + 4 * q + 0, v.x * w);
            atomAddF(dst + 4 * q + 1, v.y * w);
            atomAddF(dst + 4 * q + 2, v.z * w);
            atomAddF(dst + 4 * q + 3, v.w * w);
        }
    }
}

// plain finish (last ARMA iteration only): out = relu(acc + root + b1[k])
__global__ void finish1(const float* __restrict__ acc, const float* __restrict__ root,
                        const float* __restrict__ b1, float* __restrict__ out, int N) {
    int idx = blockIdx.x * blockDim.x + threadIdx.x;     // over K*N
    if (idx >= KST * N) return;
    int k = idx / N, n = idx % N;
    size_t base = ((size_t)k * N + n) * HCH;
#pragma unroll
    for (int f = 0; f < HCH; ++f) {
        float v = acc[base + f] + root[base + f] + b1[k * HCH + f];
        out[base + f] = fmaxf(v, 0.0f);
    }
}

// ---------------- mean over K + batchnorm ----------------
__global__ void mean_bn_stats(const float* __restrict__ out, float* __restrict__ h,
                              float* __restrict__ stats /*32: sum[16],sumsq[16]*/, int N) {
    int n = blockIdx.x * blockDim.x + threadIdx.x;
    bool act = n < N;
    float vals[HCH];
#pragma unroll
    for (int f = 0; f < HCH; ++f) vals[f] = 0.0f;
    if (act) {
        size_t b0 = (size_t)n * HCH, s = (size_t)N * HCH;
        const float inv3 = 1.0f / 3.0f;
#pragma unroll
        for (int f = 0; f < HCH; ++f)
            vals[f] = (out[b0 + f] + out[s + b0 + f] + out[2 * s + b0 + f]) * inv3;
#pragma unroll
        for (int f = 0; f < HCH; ++f) h[b0 + f] = vals[f];
    }
    int lane = threadIdx.x & 31;
#pragma unroll
    for (int f = 0; f < HCH; ++f) {
        float sv = vals[f];
        float qv = vals[f] * vals[f];
        for (int o = 16; o > 0; o >>= 1) {
            sv += __shfl_xor(sv, o, 32);
            qv += __shfl_xor(qv, o, 32);
        }
        if (lane == 0) {
            atomAddF(&stats[f], sv);
            atomAddF(&stats[HCH + f], qv);
        }
    }
}

__global__ void bn_apply(float* __restrict__ h, const float* __restrict__ stats,
                         const float* __restrict__ g, const float* __restrict__ b, int N) {
    int n = blockIdx.x * blockDim.x + threadIdx.x;
    if (n >= N) return;
    float invN = 1.0f / (float)N;
    size_t base = (size_t)n * HCH;
#pragma unroll
    for (int f = 0; f < HCH; ++f) {
        float mu  = stats[f] * invN;
        float var = stats[HCH + f] * invN - mu * mu;     // population variance (ddof=0)
        float v = (h[base + f] - mu) * rsqrtf(var + BN_EPS) * g[f] + b[f];
        h[base + f] = fmaxf(v, 0.0f);                    // ReLU
    }
}

// ---------------- ARMAConv 2 (H=16 -> C=1, act=None) ----------------
__global__ void conv2_init(const float* __restrict__ h, const float* __restrict__ w_init,
                           const float* __restrict__ w_root,
                           float* __restrict__ out2, float* __restrict__ root2, int N) {
    int idx = blockIdx.x * blockDim.x + threadIdx.x;     // over K*N
    if (idx >= KST * N) return;
    int k = idx / N, n = idx % N;
    const float* hv = h + (size_t)n * HCH;
    float a = 0.0f, r = 0.0f;
#pragma unroll
    for (int f = 0; f < HCH; ++f) {
        a += hv[f] * w_init[k * HCH + f];
        r += hv[f] * w_root[k * HCH + f];
    }
    out2[(size_t)k * N + n] = a;
    root2[(size_t)k * N + n] = r;
}

// C=1: fold out2*w2[k] (the t>0 dense transform) into the edge scale;
// thread per edge, loops over stacks
__global__ void scatter2(const float* __restrict__ out2, const int* __restrict__ ei,
                         const float* __restrict__ norm, const float* __restrict__ w2,
                         int useW, float* __restrict__ acc2, int E, int N) {
    int e = blockIdx.x * blockDim.x + threadIdx.x;
    if (e >= E) return;
    int r = ei[e], c = ei[E + e];
    float w = norm[e];
#pragma unroll
    for (int k = 0; k < KST; ++k) {
        float s = w * (useW ? w2[k] : 1.0f);
        atomAddF(&acc2[(size_t)k * N + c], out2[(size_t)k * N + r] * s);
    }
}

// out2 = acc2 + root2 + b2[k]; re-zero acc2 for next scatter
__global__ void finish2(float* __restrict__ acc2, const float* __restrict__ root2,
                        const float* __restrict__ b2, float* __restrict__ out2, int N) {
    int idx = blockIdx.x * blockDim.x + threadIdx.x;     // over K*N
    if (idx >= KST * N) return;
    int k = idx / N;
    out2[idx] = acc2[idx] + root2[idx] + b2[k];          // no activation
    acc2[idx] = 0.0f;
}

__global__ void final_sigmoid(const float* __restrict__ out2, float* __restrict__ y, int N) {
    int n = blockIdx.x * blockDim.x + threadIdx.x;
    if (n >= N) return;
    float v = (out2[n] + out2[(size_t)N + n] + out2[2 * (size_t)N + n]) * (1.0f / 3.0f);
    y[n] = 1.0f / (1.0f + expf(-v));
}

// ---------------- host orchestration ----------------
extern "C" void kernel_launch(void* const* d_in, const int* in_sizes, int n_in,
                              void* d_out, int out_size, void* d_ws, size_t ws_size,
                              hipStream_t stream) {
    (void)n_in; (void)out_size; (void)ws_size;

    const float* x       = (const float*)d_in[0];
    const int*   ei      = (const int*)  d_in[1];
    const float* ea      = (const float*)d_in[2];
    // d_in[3] = batch (unused)
    const float* w1_init = (const float*)d_in[4];
    const float* w1      = (const float*)d_in[5];
    const float* w1_root = (const float*)d_in[6];
    const float* b1      = (const float*)d_in[7];
    const float* bn_g    = (const float*)d_in[8];
    const float* bn_b    = (const float*)d_in[9];
    const float* w2_init = (const float*)d_in[10];
    const float* w2      = (const float*)d_in[11];
    const float* w2_root = (const float*)d_in[12];
    const float* b2      = (const float*)d_in[13];
    float* y = (float*)d_out;

    const int N = in_sizes[0];          // x is (N,1)
    const int E = in_sizes[2];          // edge_attr is (E,)

    // workspace carving (256B aligned)
    char* ws = (char*)d_ws;
    size_t off = 0;
    auto carve = [&](size_t bytes) -> float* {
        float* p = (float*)(ws + off);
        off += (bytes + 255) & ~(size_t)255;
        return p;
    };
    float* deg   = carve((size_t)N * 4);
    float* dinv  = carve((size_t)N * 4);
    float* norm  = carve((size_t)E * 4);
    float* out1  = carve((size_t)KST * N * HCH * 4);
    float* acc1  = carve((size_t)KST * N * HCH * 4);
    float* root1 = carve((size_t)KST * N * HCH * 4);
    float* h     = carve((size_t)N * HCH * 4);
    float* stats = carve(32 * 4);
    float* out2  = carve((size_t)KST * N * 4);
    float* acc2  = carve((size_t)KST * N * 4);
    float* root2 = carve((size_t)KST * N * 4);

    const int TPB = 256;
    auto grid = [&](long long n) { return (unsigned)((n + TPB - 1) / TPB); };

    // --- gcn_norm ---
    zero_f32<<<grid(N), TPB, 0, stream>>>(deg, N);
    deg_scatter<<<grid(E), TPB, 0, stream>>>(ei, ea, deg, E);
    compute_dinv<<<grid(N), TPB, 0, stream>>>(deg, dinv, N);
    compute_norm<<<grid(E), TPB, 0, stream>>>(ei, ea, dinv, norm, E);

    // --- ARMAConv 1 ---
    conv1_init<<<grid((long long)KST * N), TPB, 0, stream>>>(x, w1_init, w1_root, out1, root1, N);
    int numTiles = N / 16;                       // N = 100000 -> 6250 exactly
    int totalWaves = KST * numTiles;
    int wavesPerBlk = TPB / 32;
    unsigned wmmaBlocks = (unsigned)((totalWaves + wavesPerBlk - 1) / wavesPerBlk);

    zero_f32<<<grid((long long)KST * N * HCH), TPB, 0, stream>>>(acc1, KST * N * HCH);
    scatter1<<<grid(E), TPB, 0, stream>>>(out1, ei, norm, acc1, E, N);          // t = 0
    for (int t = 1; t < LIT; ++t) {
        // finish(t-1) + transform fused; also re-zeroes acc1
        arma1_fused<<<wmmaBlocks, TPB, 0, stream>>>(acc1, root1, b1, w1, out1,
                                                    numTiles, totalWaves);
        scatter1<<<grid(E), TPB, 0, stream>>>(out1, ei, norm, acc1, E, N);
    }
    finish1<<<grid((long long)KST * N), TPB, 0, stream>>>(acc1, root1, b1, out1, N);

    // --- mean over stacks + BatchNorm + ReLU ---
    zero_f32<<<1, 32, 0, stream>>>(stats, 32);
    mean_bn_stats<<<grid(N), TPB, 0, stream>>>(out1, h, stats, N);
    bn_apply<<<grid(N), TPB, 0, stream>>>(h, stats, bn_g, bn_b, N);

    // --- ARMAConv 2 ---
    conv2_init<<<grid((long long)KST * N), TPB, 0, stream>>>(h, w2_init, w2_root, out2, root2, N);
    zero_f32<<<grid((long long)KST * N), TPB, 0, stream>>>(acc2, KST * N);
    scatter2<<<grid(E), TPB, 0, stream>>>(out2, ei, norm, w2, 0, acc2, E, N);   // t = 0
    for (int t = 1; t < LIT; ++t) {
        finish2<<<grid((long long)KST * N), TPB, 0, stream>>>(acc2, root2, b2, out2, N);
        scatter2<<<grid(E), TPB, 0, stream>>>(out2, ei, norm, w2, 1, acc2, E, N);
    }
    finish2<<<grid((long long)KST * N), TPB, 0, stream>>>(acc2, root2, b2, out2, N);
    final_sigmoid<<<grid(N), TPB, 0, stream>>>(out2, y, N);
}